// SamplePatches_23545010717540
// MI455X (gfx1250) — compile-verified
//
#include <hip/hip_runtime.h>
#include <hip/hip_bf16.h>
#include <stdint.h>

// ---------------- problem constants (from reference) ----------------
#define NB        4          // batch
#define NC        3          // channels
#define HI        2048       // WSI height/width
#define GA        64         // attention grid 64x64
#define AW        (GA*GA)    // 4096 attention cells per batch
#define NP        32         // N_PATCHES
#define PATCH     128
#define TILE_EL   (PATCH*PATCH)   // 16384 elements per (b,n,c) tile
#define REF_EPS   1e-12f

typedef __attribute__((ext_vector_type(4))) unsigned int u32x4;
typedef __attribute__((ext_vector_type(8))) int          i32x8;
typedef __attribute__((ext_vector_type(4))) int          i32x4;

// ---------------- JAX-compatible Threefry-2x32 ----------------
__device__ __forceinline__ unsigned rotl32(unsigned x, int n) {
  return (x << n) | (x >> (32 - n));
}

__device__ __forceinline__ void threefry2x32(unsigned k0, unsigned k1,
                                             unsigned& x0, unsigned& x1) {
  const unsigned ks0 = k0, ks1 = k1, ks2 = 0x1BD11BEAu ^ k0 ^ k1;
  x0 += ks0; x1 += ks1;
  #define TF_R4(a,b,c,d) \
    x0 += x1; x1 = rotl32(x1,a); x1 ^= x0; \
    x0 += x1; x1 = rotl32(x1,b); x1 ^= x0; \
    x0 += x1; x1 = rotl32(x1,c); x1 ^= x0; \
    x0 += x1; x1 = rotl32(x1,d); x1 ^= x0;
  TF_R4(13,15,26,6)   x0 += ks1; x1 += ks2 + 1u;
  TF_R4(17,29,16,24)  x0 += ks2; x1 += ks0 + 2u;
  TF_R4(13,15,26,6)   x0 += ks0; x1 += ks1 + 3u;
  TF_R4(17,29,16,24)  x0 += ks1; x1 += ks2 + 4u;
  TF_R4(13,15,26,6)   x0 += ks2; x1 += ks0 + 5u;
  #undef TF_R4
}

// uniform[0,1) for flat element j of jax.random.uniform(key(42), (4,4096), f32)
// counts = iota(16384) split into halves (x0 = j, x1 = j+8192), concat outputs.
__device__ __forceinline__ float jax_uniform42(unsigned j) {
  unsigned x0, x1;
  const unsigned HALF = (NB * AW) / 2;  // 8192
  if (j < HALF) { x0 = j;        x1 = j + HALF; }
  else          { x0 = j - HALF; x1 = j;        }
  threefry2x32(0u, 42u, x0, x1);
  unsigned bits = (j < HALF) ? x0 : x1;
  bits = (bits >> 9) | 0x3F800000u;           // 23-bit mantissa in [1,2)
  return __uint_as_float(bits) - 1.0f;
}

// ---------------- kernel 1: Gumbel top-k sampling (1 block / batch) ----------------
__global__ void topk_sample_kernel(const float* __restrict__ att,
                                   float* __restrict__ att_out,
                                   int* __restrict__ yx_ws) {
  __shared__ float probs[AW];       // 16 KB
  __shared__ float pert[AW];        // 16 KB
  __shared__ float redv[256];
  __shared__ int   redi[256];

  const int b = blockIdx.x;
  const int t = threadIdx.x;
  const float* a = att + (size_t)b * AW;

  // --- sum for normalization ---
  float s = 0.0f;
  for (int i = t; i < AW; i += 256) s += a[i];
  redv[t] = s; __syncthreads();
  for (int off = 128; off > 0; off >>= 1) {
    if (t < off) redv[t] += redv[t + off];
    __syncthreads();
  }
  const float inv = 1.0f / redv[0];

  // --- probs, perturbed log-probs (logp + gumbel) ---
  for (int i = t; i < AW; i += 256) {
    float p = a[i] * inv;
    probs[i] = p;
    float u = jax_uniform42((unsigned)(b * AW + i));
    float g = -__logf(-__logf(u + REF_EPS) + REF_EPS);
    pert[i] = __logf(p + REF_EPS) + g;
  }
  __syncthreads();

  // --- iterative argmax: 32 rounds, lower-index tie-break (lax.top_k order) ---
  for (int n = 0; n < NP; n++) {
    float bv = -__builtin_inff();
    int   bi = 0x7FFFFFFF;
    for (int i = t; i < AW; i += 256) {
      float v = pert[i];
      if (v > bv || (v == bv && i < bi)) { bv = v; bi = i; }
    }
    redv[t] = bv; redi[t] = bi; __syncthreads();
    for (int off = 128; off > 0; off >>= 1) {
      if (t < off) {
        float v2 = redv[t + off]; int i2 = redi[t + off];
        if (v2 > redv[t] || (v2 == redv[t] && i2 < redi[t])) {
          redv[t] = v2; redi[t] = i2;
        }
      }
      __syncthreads();
    }
    const int m = redi[0];
    if (t == 0) {
      att_out[b * NP + n] = probs[m];
      int r = m / GA, c = m % GA;
      // sh = sw = 2048/64 = 32 ; (32-128)//2 = -48 ; clip to [0, 2048-128]
      int y0 = r * 32 - 48; y0 = min(max(y0, 0), HI - PATCH);
      int x0 = c * 32 - 48; x0 = min(max(x0, 0), HI - PATCH);
      yx_ws[(b * NP + n) * 2 + 0] = y0;
      yx_ws[(b * NP + n) * 2 + 1] = x0;
      pert[m] = -__builtin_inff();   // remove for next round (w/o replacement)
    }
    __syncthreads();
  }
}

// ---------------- kernel 2: pure-TDM tile gather (1 wave per (b,n,c) tile) ----------------
// TDM load: strided 128x128 f32 tile out of the 2048x2048 plane -> LDS
// TDM store: contiguous 16384-element tile LDS -> output
// The SIMD only assembles descriptors; all data moves on the Tensor Data Mover.
__global__ void gather_patches_tdm(const float* __restrict__ wsi,
                                   const int* __restrict__ yx_ws,
                                   float* __restrict__ patches) {
  __shared__ __align__(16) float tile[TILE_EL];   // 64 KB staging

  const int blk = blockIdx.x;        // layout (((b*NP)+n)*NC + c)
  const int c   = blk % NC;
  const int bn  = blk / NC;          // b*NP + n
  const int b   = bn / NP;

  const int y0 = __builtin_amdgcn_readfirstlane(yx_ws[bn * 2 + 0]);
  const int x0 = __builtin_amdgcn_readfirstlane(yx_ws[bn * 2 + 1]);

  const float* src = wsi + (((size_t)b * NC + c) * (size_t)HI * HI
                            + (size_t)y0 * HI + (size_t)x0);
  float* dst = patches + (size_t)blk * TILE_EL;

#if __has_builtin(__builtin_amdgcn_tensor_load_to_lds) && \
    __has_builtin(__builtin_amdgcn_tensor_store_from_lds)
  const unsigned ldsoff = (unsigned)(uintptr_t)(void*)tile;  // LDS aperture low 32 bits
  i32x4 z4 = {0, 0, 0, 0};                                   // groups 2/3 unused (2D)
#if defined(__clang_major__) && __clang_major__ >= 23
  i32x8 z8 = {0, 0, 0, 0, 0, 0, 0, 0};
#define TDM_LOAD(a, b2)  __builtin_amdgcn_tensor_load_to_lds((a), (b2), z4, z4, z8, 0)
#define TDM_STORE(a, b2) __builtin_amdgcn_tensor_store_from_lds((a), (b2), z4, z4, z8, 0)
#else
#define TDM_LOAD(a, b2)  __builtin_amdgcn_tensor_load_to_lds((a), (b2), z4, z4, 0)
#define TDM_STORE(a, b2) __builtin_amdgcn_tensor_store_from_lds((a), (b2), z4, z4, 0)
#endif

  // ---- load descriptor: 2D tensor 2048x2048 f32, tile 128x128, row stride 2048 ----
  {
    const uint64_t ga = (uint64_t)(uintptr_t)src;
    u32x4 g0;
    g0.x = 0x1u;                                   // count=1, user mode
    g0.y = ldsoff;                                 // lds_addr (bytes)
    g0.z = (unsigned)(ga & 0xFFFFFFFFu);           // global_addr[31:0]
    g0.w = (unsigned)((ga >> 32) & 0x1FFFFFFu)     // global_addr[56:32]
         | (2u << 30);                             // type = 2 ("image")
    i32x8 g1;
    g1[0] = 0x2 << 16;                             // wg_mask=0, data_size=2 (4 B)
    g1[1] = (int)((HI & 0xFFFF) << 16);            // tensor_dim0[15:0] = 2048
    g1[2] = (int)((HI & 0xFFFF) << 16);            // dim0 hi=0 | tensor_dim1[15:0]=2048
    g1[3] = (int)(PATCH << 16);                    // dim1 hi=0 | tile_dim0 = 128
    g1[4] = PATCH;                                 // tile_dim1 = 128, tile_dim2 = 0
    g1[5] = HI;                                    // tensor_dim0_stride = 2048 elems
    g1[6] = 0;                                     // stride0 hi | stride1[15:0]
    g1[7] = (int)(((uint64_t)HI * HI) >> 16);      // tensor_dim1_stride[47:16]
    TDM_LOAD(g0, g1);
  }
  __builtin_amdgcn_s_wait_tensorcnt(0);            // tile resident in LDS

  // ---- store descriptor: contiguous 16384x1 f32 tile -> output ----
  {
    const uint64_t ga = (uint64_t)(uintptr_t)dst;
    u32x4 g0;
    g0.x = 0x1u;
    g0.y = ldsoff;
    g0.z = (unsigned)(ga & 0xFFFFFFFFu);
    g0.w = (unsigned)((ga >> 32) & 0x1FFFFFFu) | (2u << 30);
    i32x8 g1;
    g1[0] = 0x2 << 16;                             // data_size = 4 B
    g1[1] = (int)((TILE_EL & 0xFFFF) << 16);       // tensor_dim0[15:0] = 16384
    g1[2] = (int)(1 << 16);                        // dim0 hi=0 | tensor_dim1 = 1
    g1[3] = (int)((TILE_EL & 0xFFFF) << 16);       // dim1 hi=0 | tile_dim0 = 16384
    g1[4] = 1;                                     // tile_dim1 = 1, tile_dim2 = 0
    g1[5] = TILE_EL;                               // tensor_dim0_stride = 16384
    g1[6] = 0;
    g1[7] = 0;
    TDM_STORE(g0, g1);
  }
  __builtin_amdgcn_s_wait_tensorcnt(0);            // store complete before endpgm
#undef TDM_LOAD
#undef TDM_STORE
#else
  // fallback (no TDM builtins): direct strided global->global copy
  for (int i = threadIdx.x; i < TILE_EL; i += blockDim.x)
    dst[i] = src[(size_t)(i >> 7) * HI + (i & (PATCH - 1))];
  (void)tile;
#endif
}

// ---------------- host-side launcher ----------------
extern "C" void kernel_launch(void* const* d_in, const int* in_sizes, int n_in,
                              void* d_out, int out_size, void* d_ws, size_t ws_size,
                              hipStream_t stream) {
  // setup_inputs order: x_low[0], x_high[1], attention[2], WSI[3]
  const float* attention = (const float*)d_in[2];
  const float* wsi       = (const float*)d_in[3];

  float* out      = (float*)d_out;
  float* patches  = out;                                         // [4,32,3,128,128]
  float* att_out  = out + (size_t)NB * NP * NC * PATCH * PATCH;  // [4,32]
  int*   yx_ws    = (int*)d_ws;                                  // [4*32][2]

  topk_sample_kernel<<<dim3(NB), dim3(256), 0, stream>>>(attention, att_out, yx_ws);
  // one wave per tile: descriptors only, all data moves on the TDM engines
  gather_patches_tdm<<<dim3(NB * NP * NC), dim3(32), 0, stream>>>(wsi, yx_ws, patches);
}